// NexusNet_13846974562668
// MI455X (gfx1250) — compile-verified
//
#include <hip/hip_runtime.h>
#include <math.h>

#define P_ 3
#define N_ 100000
#define M_ 30000
#define E_ 200000
#define C_ 5
#define FP_ 64
#define FN_ 32

typedef __attribute__((ext_vector_type(2))) float v2f;
typedef __attribute__((ext_vector_type(8))) float v8f;
typedef __attribute__((ext_vector_type(4))) unsigned u32x4;
typedef __attribute__((ext_vector_type(4))) int i32x4;
typedef __attribute__((ext_vector_type(8))) int i32x8;

// D = A(16x4 f32) * B(4x16 f32) + C, full-precision f32 WMMA
__device__ __forceinline__ v8f wmma4(v2f a, v2f b, v8f c) {
    return __builtin_amdgcn_wmma_f32_16x16x4_f32(false, a, false, b, (short)0, c,
                                                 false, false);
}

// LDS anti-bank-conflict padding: +2 dwords every 32 dwords.  Matches the
// TDM descriptor's pad_enable/pad_interval=32/pad_amount=2 hardware padding.
// For even idx the padded offset stays even -> ds_load_b64 stays 8B aligned.
__device__ __forceinline__ int padIdx(int idx) { return idx + ((idx >> 5) << 1); }
#define PADSZ(n) ((n) + (((n) >> 5) << 1))

// ---------------------------------------------------------------------------
// Tensor Data Mover: 1-D tile of nElems f32 from global -> LDS, with
// hardware padding (2 dwords per 32).  Issued once per wave (EXEC ignored).
// D# per CDNA5 ISA.
// ---------------------------------------------------------------------------
__device__ __forceinline__ void tdm_load_f32(const float* gsrc, float* ldst,
                                             unsigned nElems) {
    unsigned lds_addr = (unsigned)(size_t)(void*)ldst;     // low 32 bits = LDS offset
    unsigned long long ga = (unsigned long long)(size_t)gsrc;

    u32x4 g0;
    g0.x = 1u;                                             // count=1 valid user D#
    g0.y = lds_addr;                                       // lds_addr
    g0.z = (unsigned)ga;                                   // global_addr[31:0]
    g0.w = (unsigned)((ga >> 32) & 0x01FFFFFFull) | (2u << 30); // addr[56:32] | type=2

    i32x8 g1;
    // data_size=2(4B) | pad_enable | pad_interval=4(32 dwords) | pad_amount=1(2 dwords)
    g1.s0 = (int)((2u << 16) | (1u << 20) | (4u << 22) | (1u << 25));
    g1.s1 = (int)((nElems & 0xFFFFu) << 16);               // tensor_dim0[15:0] @63:48
    g1.s2 = (int)(((nElems >> 16) & 0xFFFFu) | (1u << 16)); // dim0 hi | tensor_dim1=1
    g1.s3 = (int)((nElems & 0xFFFFu) << 16);               // tile_dim0 @127:112
    g1.s4 = 0;                                             // tile_dim1=0, tile_dim2=0
    g1.s5 = (int)nElems;                                   // tensor_dim0_stride lo
    g1.s6 = 0;
    g1.s7 = 0;

    i32x4 z4 = {0, 0, 0, 0};
#if __clang_major__ >= 23
    i32x8 z8 = {0, 0, 0, 0, 0, 0, 0, 0};
    __builtin_amdgcn_tensor_load_to_lds(g0, g1, z4, z4, z8, 0);
#else
    __builtin_amdgcn_tensor_load_to_lds(g0, g1, z4, z4, 0);
#endif
}

// ---------------------------------------------------------------------------
// Kernel 1: nexus_up scatter.  One block per (plane, edge); 320 threads cover
// C*FP = 320 features; accumulate into concat layout n_cat[M, C, P*FP].
// ---------------------------------------------------------------------------
__global__ void scatter_up_kernel(const float* __restrict__ x,     // [P,N,C,FP]
                                  const int* __restrict__ esrc,    // [P,E]
                                  const int* __restrict__ edst,    // [P,E]
                                  float* __restrict__ ncat)        // [M,C,P*FP]
{
    unsigned pe = blockIdx.x;
    int p = pe / E_;
    int e = pe % E_;
    int s = esrc[p * E_ + e];
    int d = edst[p * E_ + e];
    int t = threadIdx.x;            // 0..319
    int c = t >> 6;
    int f = t & 63;
    float v = x[(((size_t)p * N_ + s) * C_ + c) * FP_ + f];
    atomicAdd(&ncat[((size_t)d * C_ + c) * (P_ * FP_) + p * FP_ + f], v);
}

// ---------------------------------------------------------------------------
// Kernel 2: nexus_net.  Block = 4 waves, all the SAME class; class weights
// (Wn1 24KB + Wn2 4KB) staged to LDS once per block via TDM; each wave does a
// 16-row tile: [16,192]x[192,32] then [16,32]x[32,32], WMMA f32 16x16x4.
// ---------------------------------------------------------------------------
__global__ void nexus_net_kernel(const float* __restrict__ ncat,  // [M,C,192]
                                 const float* __restrict__ Wn1,   // [C,32,192]
                                 const float* __restrict__ bn1,   // [C,32]
                                 const float* __restrict__ Wn2,   // [C,32,32]
                                 const float* __restrict__ bn2,   // [C,32]
                                 float* __restrict__ nout)        // [M,C,32]
{
    const int lane = threadIdx.x & 31;
    const int wib  = threadIdx.x >> 5;
    const int nTiles = M_ / 16;                       // 1875
    const int blocksPerClass = (nTiles + 3) / 4;      // 469
    const int c  = blockIdx.x / blocksPerClass;
    const int tg = blockIdx.x % blocksPerClass;
    int tile = tg * 4 + wib;
    tile = tile < nTiles ? tile : nTiles - 1;         // clamp: EXEC stays full
    const int m0 = tile * 16;
    const int row = lane & 15, half = lane >> 4, col = lane & 15;

    __shared__ __align__(16) float w1_s[PADSZ(32 * 192)];
    __shared__ __align__(16) float w2_s[PADSZ(32 * 32)];
    __shared__ __align__(16) float h_s[4][PADSZ(16 * 32)];
    float* hbuf = h_s[wib];

    if (wib == 0) {
        tdm_load_f32(Wn1 + (size_t)c * 32 * 192, w1_s, 32 * 192);
        tdm_load_f32(Wn2 + (size_t)c * 32 * 32,  w2_s, 32 * 32);
        __builtin_amdgcn_s_wait_tensorcnt(0);
    }
    __syncthreads();

    // ---- layer 1: [16,192] x [192,32] ----
    v8f a0 = {}, a1 = {};
    const float* arow = ncat + ((size_t)(m0 + row) * C_ + c) * (P_ * FP_);
    for (int kb = 0; kb < 192; kb += 4) {
        int k = kb + 2 * half;
        v2f a  = *(const v2f*)(arow + k);
        v2f b0 = *(const v2f*)(w1_s + padIdx(col * 192 + k));
        v2f b1 = *(const v2f*)(w1_s + padIdx((16 + col) * 192 + k));
        a0 = wmma4(a, b0, a0);
        a1 = wmma4(a, b1, a1);
    }
    {
        const float* bb = bn1 + c * 32;
        for (int r = 0; r < 8; ++r) {
            int rr = r + 8 * half;
            hbuf[padIdx(rr * 32 + col)]      = tanhf(a0[r] + bb[col]);
            hbuf[padIdx(rr * 32 + 16 + col)] = tanhf(a1[r] + bb[16 + col]);
        }
    }
    // same-wave DS ops are in-order -> no barrier needed for per-wave hbuf

    // ---- layer 2: [16,32] x [32,32] ----
    v8f d0 = {}, d1 = {};
    for (int kb = 0; kb < 32; kb += 4) {
        int k = kb + 2 * half;
        v2f a  = *(const v2f*)(hbuf + padIdx(row * 32 + k));
        v2f b0 = *(const v2f*)(w2_s + padIdx(col * 32 + k));
        v2f b1 = *(const v2f*)(w2_s + padIdx((16 + col) * 32 + k));
        d0 = wmma4(a, b0, d0);
        d1 = wmma4(a, b1, d1);
    }
    {
        const float* bb = bn2 + c * 32;
        for (int r = 0; r < 8; ++r) {
            int rr = r + 8 * half;
            float* orow = nout + ((size_t)(m0 + rr) * C_ + c) * FN_;
            orow[col]      = tanhf(d0[r] + bb[col]);
            orow[16 + col] = tanhf(d1[r] + bb[16 + col]);
        }
    }
}

// ---------------------------------------------------------------------------
// Kernel 3: per-plane edge attention + message scatter (sum + count).
// One wave32 per edge; lanes split the 96-feature dots; shuffle reduce;
// softmax over C=5 computed redundantly per lane.
// ---------------------------------------------------------------------------
__global__ void edge_kernel(const float* __restrict__ xp,    // [N,C,FP]
                            const float* __restrict__ nex,   // [M,C,FN]
                            const int* __restrict__ src,     // [E]
                            const int* __restrict__ dst,     // [E]
                            const float* __restrict__ Wep,   // [C,96]
                            const float* __restrict__ bep,   // [C]
                            float* __restrict__ aggr,        // [N,C,FN] sums
                            float* __restrict__ cnt)         // [N]
{
    const int lane = threadIdx.x & 31;
    const int e = blockIdx.x * (blockDim.x >> 5) + (threadIdx.x >> 5);
    if (e >= E_) return;
    const int s = src[e], d = dst[e];
    const float* xi = xp  + (size_t)s * (C_ * FP_);
    const float* nj = nex + (size_t)d * (C_ * FN_);

    float logits[C_];
    for (int c = 0; c < C_; ++c) {
        const float* w = Wep + c * 96;
        float part = w[lane]      * xi[c * FP_ + lane]
                   + w[lane + 32] * xi[c * FP_ + lane + 32]
                   + w[lane + 64] * nj[c * FN_ + lane];
        for (int off = 16; off; off >>= 1)
            part += __shfl_down(part, off, 32);
        part = __shfl(part, 0, 32);
        logits[c] = part + bep[c];
    }
    float mx = logits[0];
    for (int c = 1; c < C_; ++c) mx = fmaxf(mx, logits[c]);
    float wgt[C_], sum = 0.f;
    for (int c = 0; c < C_; ++c) { wgt[c] = __expf(logits[c] - mx); sum += wgt[c]; }
    const float inv = 1.f / sum;

    for (int c = 0; c < C_; ++c)
        atomicAdd(&aggr[((size_t)s * C_ + c) * FN_ + lane],
                  wgt[c] * inv * nj[c * FN_ + lane]);
    if (lane == 0) atomicAdd(&cnt[s], 1.0f);
}

// ---------------------------------------------------------------------------
// Kernel 4: per-plane down-net.  Block = 4 waves of one class; class weights
// (Wd1 24KB + Wd2 16KB) staged via TDM once per block; per wave:
//   feat[16,96] = concat(x, aggr/cnt) staged in padded LDS,
//   layer1 [16,96]x[96,64] (96 wmma), layer2 [16,64]x[64,64] (64 wmma),
//   h aliases the feat buffer (dead after layer1 K-loop).
// ---------------------------------------------------------------------------
__global__ void down_kernel(const float* __restrict__ xp,    // [N,C,FP]
                            const float* __restrict__ aggr,  // [N,C,FN]
                            const float* __restrict__ cnt,   // [N]
                            const float* __restrict__ W1,    // [C,64,96]
                            const float* __restrict__ b1,    // [C,64]
                            const float* __restrict__ W2,    // [C,64,64]
                            const float* __restrict__ b2,    // [C,64]
                            float* __restrict__ outp)        // [N,C,FP]
{
    const int lane = threadIdx.x & 31;
    const int wib  = threadIdx.x >> 5;
    const int nTiles = N_ / 16;                        // 6250
    const int blocksPerClass = (nTiles + 3) / 4;       // 1563
    const int c  = blockIdx.x / blocksPerClass;
    const int tg = blockIdx.x % blocksPerClass;
    int tile = tg * 4 + wib;
    tile = tile < nTiles ? tile : nTiles - 1;          // clamp: EXEC stays full
    const int n0 = tile * 16;
    const int row = lane & 15, half = lane >> 4, col = lane & 15;

    __shared__ __align__(16) float w1_s[PADSZ(64 * 96)];
    __shared__ __align__(16) float w2_s[PADSZ(64 * 64)];
    __shared__ __align__(16) float feat_s[4][PADSZ(16 * 96)];
    float* feat = feat_s[wib];
    float* hbuf = feat;                                // alias (feat dead after L1)

    if (wib == 0) {
        tdm_load_f32(W1 + (size_t)c * 64 * 96, w1_s, 64 * 96);
        tdm_load_f32(W2 + (size_t)c * 64 * 64, w2_s, 64 * 64);
        __builtin_amdgcn_s_wait_tensorcnt(0);
    }
    __syncthreads();

    // stage feat[16][96]: cols 0..63 = x, 64..95 = aggr / max(cnt,1)
    for (int i = lane; i < 16 * 96; i += 32) {
        int r = i / 96, f = i % 96;
        int node = n0 + r;
        float v;
        if (f < 64) {
            v = xp[((size_t)node * C_ + c) * FP_ + f];
        } else {
            float ct = fmaxf(cnt[node], 1.0f);
            v = aggr[((size_t)node * C_ + c) * FN_ + (f - 64)] / ct;
        }
        feat[padIdx(i)] = v;
    }

    // ---- layer 1: [16,96] x [96,64], 4 column tiles ----
    v8f a0 = {}, a1 = {}, a2 = {}, a3 = {};
    for (int kb = 0; kb < 96; kb += 4) {
        int k = kb + 2 * half;
        v2f a   = *(const v2f*)(feat + padIdx(row * 96 + k));
        v2f b0v = *(const v2f*)(w1_s + padIdx((col)      * 96 + k));
        v2f b1v = *(const v2f*)(w1_s + padIdx((16 + col) * 96 + k));
        v2f b2v = *(const v2f*)(w1_s + padIdx((32 + col) * 96 + k));
        v2f b3v = *(const v2f*)(w1_s + padIdx((48 + col) * 96 + k));
        a0 = wmma4(a, b0v, a0);
        a1 = wmma4(a, b1v, a1);
        a2 = wmma4(a, b2v, a2);
        a3 = wmma4(a, b3v, a3);
    }
    {
        const float* bb = b1 + c * 64;
        for (int r = 0; r < 8; ++r) {
            int rr = r + 8 * half;
            hbuf[padIdx(rr * 64 + col)]      = tanhf(a0[r] + bb[col]);
            hbuf[padIdx(rr * 64 + 16 + col)] = tanhf(a1[r] + bb[16 + col]);
            hbuf[padIdx(rr * 64 + 32 + col)] = tanhf(a2[r] + bb[32 + col]);
            hbuf[padIdx(rr * 64 + 48 + col)] = tanhf(a3[r] + bb[48 + col]);
        }
    }

    // ---- layer 2: [16,64] x [64,64], 4 column tiles ----
    v8f d0 = {}, d1 = {}, d2 = {}, d3 = {};
    for (int kb = 0; kb < 64; kb += 4) {
        int k = kb + 2 * half;
        v2f a   = *(const v2f*)(hbuf + padIdx(row * 64 + k));
        v2f b0v = *(const v2f*)(w2_s + padIdx((col)      * 64 + k));
        v2f b1v = *(const v2f*)(w2_s + padIdx((16 + col) * 64 + k));
        v2f b2v = *(const v2f*)(w2_s + padIdx((32 + col) * 64 + k));
        v2f b3v = *(const v2f*)(w2_s + padIdx((48 + col) * 64 + k));
        d0 = wmma4(a, b0v, d0);
        d1 = wmma4(a, b1v, d1);
        d2 = wmma4(a, b2v, d2);
        d3 = wmma4(a, b3v, d3);
    }
    {
        const float* bb = b2 + c * 64;
        for (int r = 0; r < 8; ++r) {
            int rr = r + 8 * half;
            float* orow = outp + ((size_t)(n0 + rr) * C_ + c) * FP_;
            orow[col]      = tanhf(d0[r] + bb[col]);
            orow[16 + col] = tanhf(d1[r] + bb[16 + col]);
            orow[32 + col] = tanhf(d2[r] + bb[32 + col]);
            orow[48 + col] = tanhf(d3[r] + bb[48 + col]);
        }
    }
}

// ---------------------------------------------------------------------------
extern "C" void kernel_launch(void* const* d_in, const int* in_sizes, int n_in,
                              void* d_out, int out_size, void* d_ws, size_t ws_size,
                              hipStream_t stream) {
    const float* x    = (const float*)d_in[0];   // [P,N,C,FP]
    // d_in[1] = nexus (unused by reference)
    const int*   esrc = (const int*)d_in[2];     // [P,E]
    const int*   edst = (const int*)d_in[3];     // [P,E]
    const float* Wn1  = (const float*)d_in[4];   // [C,32,192]
    const float* bn1  = (const float*)d_in[5];   // [C,32]
    const float* Wn2  = (const float*)d_in[6];   // [C,32,32]
    const float* bn2  = (const float*)d_in[7];   // [C,32]
    const float* We   = (const float*)d_in[8];   // [P,C,1,96]
    const float* be   = (const float*)d_in[9];   // [P,C,1]
    const float* Wd1  = (const float*)d_in[10];  // [P,C,64,96]
    const float* bd1  = (const float*)d_in[11];  // [P,C,64]
    const float* Wd2  = (const float*)d_in[12];  // [P,C,64,64]
    const float* bd2  = (const float*)d_in[13];  // [P,C,64]
    float*       out  = (float*)d_out;           // [P,N,C,FP]

    // workspace layout (floats)
    float* ncat = (float*)d_ws;                              // M*C*192
    float* nex  = ncat + (size_t)M_ * C_ * (P_ * FP_);       // M*C*32
    float* aggr = nex  + (size_t)M_ * C_ * FN_;              // N*C*32 (per-plane reuse)
    float* cnt  = aggr + (size_t)N_ * C_ * FN_;              // N

    hipMemsetAsync(ncat, 0, (size_t)M_ * C_ * (P_ * FP_) * sizeof(float), stream);

    // 1) nexus_up edge scatter
    scatter_up_kernel<<<P_ * E_, 320, 0, stream>>>(x, esrc, edst, ncat);

    // 2) nexus_net (TDM weight staging + WMMA)
    {
        int blocksPerClass = (M_ / 16 + 3) / 4;
        nexus_net_kernel<<<C_ * blocksPerClass, 128, 0, stream>>>(
            ncat, Wn1, bn1, Wn2, bn2, nex);
    }

    // 3) per-plane: edge softmax scatter-mean, then down-net
    for (int p = 0; p < P_; ++p) {
        hipMemsetAsync(aggr, 0,
                       ((size_t)N_ * C_ * FN_ + N_) * sizeof(float), stream);

        {
            int wavesPerBlock = 8;
            dim3 grid((E_ + wavesPerBlock - 1) / wavesPerBlock);
            edge_kernel<<<grid, 32 * wavesPerBlock, 0, stream>>>(
                x + (size_t)p * N_ * C_ * FP_, nex,
                esrc + (size_t)p * E_, edst + (size_t)p * E_,
                We + (size_t)p * C_ * 96, be + (size_t)p * C_,
                aggr, cnt);
        }
        {
            int blocksPerClass = (N_ / 16 + 3) / 4;
            down_kernel<<<C_ * blocksPerClass, 128, 0, stream>>>(
                x + (size_t)p * N_ * C_ * FP_, aggr, cnt,
                Wd1 + (size_t)p * C_ * 64 * 96, bd1 + (size_t)p * C_ * 64,
                Wd2 + (size_t)p * C_ * 64 * 64, bd2 + (size_t)p * C_ * 64,
                out + (size_t)p * N_ * C_ * FP_);
        }
    }
}